// GATBasedMolecularGraphNeuralNetwork_54872502173941
// MI455X (gfx1250) — compile-verified
//
#include <hip/hip_runtime.h>
#include <hip/hip_bf16.h>

typedef __attribute__((ext_vector_type(2))) float v2f;
typedef __attribute__((ext_vector_type(8))) float v8f;

// ---------- helpers: order-preserving float<->uint for atomic max ----------
__device__ __forceinline__ unsigned f2ord(float f) {
    unsigned u = __float_as_uint(f);
    return (u & 0x80000000u) ? ~u : (u | 0x80000000u);
}
__device__ __forceinline__ float ord2f(unsigned k) {
    return (k & 0x80000000u) ? __uint_as_float(k & 0x7fffffffu)
                             : __uint_as_float(~k);
}

// ---------- build h0 = [x(3), emb[atom](10), 0,0,0]  (N x 16 padded) ----------
__global__ void build_h0_kernel(const float* __restrict__ x,
                                const float* __restrict__ emb,
                                float* __restrict__ h0, int N) {
    int n = blockIdx.x * blockDim.x + threadIdx.x;
    if (n >= N) return;
    float x0 = x[3 * (long long)n + 0];
    float x1 = x[3 * (long long)n + 1];
    float x2 = x[3 * (long long)n + 2];
    int atom = (int)x0;
    float* o = h0 + (long long)n * 16;
    o[0] = x0; o[1] = x1; o[2] = x2;
#pragma unroll
    for (int j = 0; j < 10; ++j) o[3 + j] = emb[atom * 10 + j];
    o[13] = 0.f; o[14] = 0.f; o[15] = 0.f;
}

// ---------- fp32 WMMA GEMM:  XL = act(H) @ Wl,  XR = act(H) @ Wr ----------
// H: N x ldh (row major, Kpad <= ldh cols valid/padded), W: Klog x 64.
// 128 threads = 4 waves; wave w handles rows [blk*64 + w*16, +16), all 64 cols.
__global__ void __launch_bounds__(128)
gemm_xlxr_kernel(const float* __restrict__ H, int ldh, int Kpad, int Klog, int N,
                 const float* __restrict__ Wl, const float* __restrict__ Wr,
                 float slope,
                 float* __restrict__ XL, float* __restrict__ XR) {
    __shared__ float sWl[64 * 64];
    __shared__ float sWr[64 * 64];
    // stage weights (zero-padded to Kpad rows) into LDS
    for (int idx = threadIdx.x; idx < Kpad * 64; idx += blockDim.x) {
        int k = idx >> 6, n = idx & 63;
        sWl[idx] = (k < Klog) ? Wl[k * 64 + n] : 0.f;
        sWr[idx] = (k < Klog) ? Wr[k * 64 + n] : 0.f;
    }
    __syncthreads();

    const int lane = threadIdx.x & 31;
    const int wave = threadIdx.x >> 5;
    const int hi   = lane >> 4;        // 0: K pair {0,1}, 1: K pair {2,3}
    const int l16  = lane & 15;        // row (A) / col (B,C,D) within tile
    const long long rowBase = (long long)blockIdx.x * 64 + wave * 16;

    v8f accL[4] = {v8f{}, v8f{}, v8f{}, v8f{}};
    v8f accR[4] = {v8f{}, v8f{}, v8f{}, v8f{}};

    for (int k = 0; k < Kpad; k += 4) {
        int ka = k + hi * 2;
        long long row = rowBase + l16;
        if (row >= N) row = N - 1;          // keep EXEC full; stores predicated
        float a0 = H[row * ldh + ka];
        float a1 = H[row * ldh + ka + 1];
        if (slope != 0.f) {                 // fused leaky_relu on GEMM input
            a0 = a0 > 0.f ? a0 : slope * a0;
            a1 = a1 > 0.f ? a1 : slope * a1;
        }
        v2f a; a.x = a0; a.y = a1;
#pragma unroll
        for (int t = 0; t < 4; ++t) {
            v2f bl, br;
            bl.x = sWl[ka * 64 + t * 16 + l16];
            bl.y = sWl[(ka + 1) * 64 + t * 16 + l16];
            br.x = sWr[ka * 64 + t * 16 + l16];
            br.y = sWr[(ka + 1) * 64 + t * 16 + l16];
            accL[t] = __builtin_amdgcn_wmma_f32_16x16x4_f32(
                false, a, false, bl, (short)0, accL[t], false, false);
            accR[t] = __builtin_amdgcn_wmma_f32_16x16x4_f32(
                false, a, false, br, (short)0, accR[t], false, false);
        }
    }

#pragma unroll
    for (int t = 0; t < 4; ++t)
#pragma unroll
        for (int r = 0; r < 8; ++r) {
            long long m = rowBase + hi * 8 + r;
            if (m < N) {
                XL[m * 64 + t * 16 + l16] = accL[t][r];
                XR[m * 64 + t * 16 + l16] = accR[t][r];
            }
        }
}

// ---------- init out buffer with bias (out = b broadcast over nodes) ----------
__global__ void init_bias_kernel(float* __restrict__ out,
                                 const float* __restrict__ bias, long long total) {
    long long i = (long long)blockIdx.x * blockDim.x + threadIdx.x;
    if (i < total) out[i] = bias[i & 63];
}

// ---------- edge logits: e = leaky(xl[s]+xr[d],0.2) . att ; segment max ----------
__global__ void edge_logits_kernel(const int* __restrict__ ei, int E, int Etot,
                                   const float* __restrict__ xl,
                                   const float* __restrict__ xr,
                                   const float* __restrict__ att,
                                   float* __restrict__ ebuf,
                                   unsigned* __restrict__ mOrd) {
    int wid  = (blockIdx.x * blockDim.x + threadIdx.x) >> 5;
    int lane = threadIdx.x & 31;
    if (wid >= Etot) return;
    int s, d;
    if (wid < E) { s = ei[wid]; d = ei[E + wid]; } else { s = d = wid - E; }
    float v0 = xl[(long long)s * 64 + lane]      + xr[(long long)d * 64 + lane];
    float v1 = xl[(long long)s * 64 + 32 + lane] + xr[(long long)d * 64 + 32 + lane];
    v0 = v0 > 0.f ? v0 : 0.2f * v0;
    v1 = v1 > 0.f ? v1 : 0.2f * v1;
    float e = v0 * att[lane] + v1 * att[32 + lane];
#pragma unroll
    for (int o = 16; o; o >>= 1) e += __shfl_xor(e, o, 32);
    if (lane == 0) {
        ebuf[wid] = e;
        atomicMax(&mOrd[d], f2ord(e));
    }
}

// ---------- softmax numerator + segment sum ----------
__global__ void edge_softmax_kernel(const int* __restrict__ ei, int E, int Etot,
                                    const unsigned* __restrict__ mOrd,
                                    float* __restrict__ ebuf,
                                    float* __restrict__ ssum) {
    int i = blockIdx.x * blockDim.x + threadIdx.x;
    if (i >= Etot) return;
    int d = (i < E) ? ei[E + i] : (i - E);
    float p = __expf(ebuf[i] - ord2f(mOrd[d]));
    ebuf[i] = p;
    atomicAdd(&ssum[d], p);
}

// ---------- weighted scatter: out[d] += (p/s[d]) * xl[s] ----------
__global__ void edge_scatter_kernel(const int* __restrict__ ei, int E, int Etot,
                                    const float* __restrict__ ebuf,
                                    const float* __restrict__ ssum,
                                    const float* __restrict__ xl,
                                    float* __restrict__ out) {
    int wid  = (blockIdx.x * blockDim.x + threadIdx.x) >> 5;
    int lane = threadIdx.x & 31;
    if (wid >= Etot) return;
    int s, d;
    if (wid < E) { s = ei[wid]; d = ei[E + wid]; } else { s = d = wid - E; }
    float w = ebuf[wid] / fmaxf(ssum[d], 1e-16f);
    atomicAdd(&out[(long long)d * 64 + lane],      w * xl[(long long)s * 64 + lane]);
    atomicAdd(&out[(long long)d * 64 + 32 + lane], w * xl[(long long)s * 64 + 32 + lane]);
}

// ---------- graph pooling: mean-sum, max, count ----------
__global__ void pool_kernel(const float* __restrict__ h,
                            const int* __restrict__ batch, int N,
                            float* __restrict__ psum,
                            unsigned* __restrict__ pmOrd,
                            float* __restrict__ cnt) {
    int node = (blockIdx.x * blockDim.x + threadIdx.x) >> 5;
    int lane = threadIdx.x & 31;
    if (node >= N) return;
    int g = batch[node];
    float a = h[(long long)node * 64 + lane];
    float b = h[(long long)node * 64 + 32 + lane];
    atomicAdd(&psum[(long long)g * 64 + lane], a);
    atomicAdd(&psum[(long long)g * 64 + 32 + lane], b);
    atomicMax(&pmOrd[(long long)g * 64 + lane], f2ord(a));
    atomicMax(&pmOrd[(long long)g * 64 + 32 + lane], f2ord(b));
    if (lane == 0) atomicAdd(&cnt[g], 1.0f);
}

// ---------- final MLP: z(131) -> 64 (leaky 0.01) -> 1 ; block per graph ----------
__global__ void __launch_bounds__(64)
final_mlp_kernel(const float* __restrict__ psum, const unsigned* __restrict__ pmOrd,
                 const float* __restrict__ cnt, const float* __restrict__ protein,
                 const float* __restrict__ fc_w, const float* __restrict__ fc_b,
                 const float* __restrict__ fc2_w, const float* __restrict__ fc2_b,
                 float* __restrict__ out, int G) {
    int g = blockIdx.x;
    int j = threadIdx.x;           // 64 threads
    __shared__ float z[131];
    __shared__ float hbuf[64];
    float c = cnt[g];
    for (int k = j; k < 131; k += 64) {
        float v;
        if (k < 64)       v = psum[(long long)g * 64 + k] / fmaxf(c, 1.0f);
        else if (k < 67)  v = protein[(long long)g * 3 + (k - 64)];
        else              v = (c > 0.f) ? ord2f(pmOrd[(long long)g * 64 + (k - 67)]) : 0.f;
        z[k] = v;
    }
    __syncthreads();
    float acc = fc_b[j];
    for (int k = 0; k < 131; ++k) acc += z[k] * fc_w[k * 64 + j];
    acc = acc > 0.f ? acc : 0.01f * acc;
    hbuf[j] = acc * fc2_w[j];
    __syncthreads();
    if (j == 0) {
        float s = 0.f;
#pragma unroll
        for (int k = 0; k < 64; ++k) s += hbuf[k];
        out[g] = s + fc2_b[0];
    }
}

// ============================ host orchestration ============================
extern "C" void kernel_launch(void* const* d_in, const int* in_sizes, int n_in,
                              void* d_out, int out_size, void* d_ws, size_t ws_size,
                              hipStream_t stream) {
    const float* x      = (const float*)d_in[0];
    const int*   ei     = (const int*)  d_in[1];
    const int*   batch  = (const int*)  d_in[2];
    const float* prot   = (const float*)d_in[3];
    const float* emb    = (const float*)d_in[4];
    const float* Wl0    = (const float*)d_in[5];
    const float* Wr0    = (const float*)d_in[6];
    const float* att0   = (const float*)d_in[7];
    const float* b0     = (const float*)d_in[8];
    const float* Wl     = (const float*)d_in[9];   // 5 x 64 x 64
    const float* Wr     = (const float*)d_in[10];
    const float* att    = (const float*)d_in[11];  // 5 x 64
    const float* bb     = (const float*)d_in[12];  // 5 x 64
    const float* fc_w   = (const float*)d_in[13];
    const float* fc_b   = (const float*)d_in[14];
    const float* fc2_w  = (const float*)d_in[15];
    const float* fc2_b  = (const float*)d_in[16];
    float* out = (float*)d_out;

    const int N    = in_sizes[0] / 3;
    const int E    = in_sizes[1] / 2;
    const int G    = in_sizes[3] / 3;
    const int Etot = E + N;

    // workspace carve-up (256B aligned)
    char*  ws  = (char*)d_ws;
    size_t off = 0;
    auto carve = [&](size_t bytes) -> void* {
        void* p = ws + off;
        off = (off + bytes + 255) & ~(size_t)255;
        return p;
    };
    float*    h0    = (float*)   carve((size_t)N * 16 * 4);
    float*    hA    = (float*)   carve((size_t)N * 64 * 4);
    float*    hB    = (float*)   carve((size_t)N * 64 * 4);
    float*    xl    = (float*)   carve((size_t)N * 64 * 4);
    float*    xr    = (float*)   carve((size_t)N * 64 * 4);
    float*    ebuf  = (float*)   carve((size_t)Etot * 4);
    unsigned* mOrd  = (unsigned*)carve((size_t)N * 4);
    float*    ssum  = (float*)   carve((size_t)N * 4);
    float*    psum  = (float*)   carve((size_t)G * 64 * 4);
    unsigned* pmOrd = (unsigned*)carve((size_t)G * 64 * 4);
    float*    cnt   = (float*)   carve((size_t)G * 4);

    const int T = 256;
    const int gridNodeThreads = (N + T - 1) / T;
    const int gridEdgeWaves   = ((Etot * 32) + T - 1) / T;   // wave per edge
    const int gridEdgeThreads = (Etot + T - 1) / T;
    const int gridNodeWaves   = ((N * 32) + T - 1) / T;      // wave per node
    const int gridGemm        = (N + 63) / 64;

    // h0 = [x, emb[atom]] padded to 16
    build_h0_kernel<<<gridNodeThreads, T, 0, stream>>>(x, emb, h0, N);

    // 6 GATv2 layers
    for (int layer = 0; layer < 6; ++layer) {
        const float* hin;  int ldh, Kpad, Klog;  float slope;
        const float* wl; const float* wr; const float* a; const float* bias;
        float* hout;
        if (layer == 0) {
            hin = h0; ldh = 16; Kpad = 16; Klog = 13; slope = 0.f;
            wl = Wl0; wr = Wr0; a = att0; bias = b0;
            hout = hA;
        } else {
            int i = layer - 1;
            hin  = (layer & 1) ? hA : hB;       // 1:hA 2:hB 3:hA 4:hB 5:hA
            hout = (layer & 1) ? hB : hA;
            ldh = 64; Kpad = 64; Klog = 64; slope = 0.01f;
            wl = Wl + (size_t)i * 64 * 64; wr = Wr + (size_t)i * 64 * 64;
            a  = att + (size_t)i * 64;     bias = bb + (size_t)i * 64;
        }

        gemm_xlxr_kernel<<<gridGemm, 128, 0, stream>>>(
            hin, ldh, Kpad, Klog, N, wl, wr, slope, xl, xr);

        hipMemsetAsync(mOrd, 0, (size_t)N * 4, stream);
        hipMemsetAsync(ssum, 0, (size_t)N * 4, stream);
        init_bias_kernel<<<(int)(((size_t)N * 64 + T - 1) / T), T, 0, stream>>>(
            hout, bias, (long long)N * 64);

        edge_logits_kernel<<<gridEdgeWaves, T, 0, stream>>>(
            ei, E, Etot, xl, xr, a, ebuf, mOrd);
        edge_softmax_kernel<<<gridEdgeThreads, T, 0, stream>>>(
            ei, E, Etot, mOrd, ebuf, ssum);
        edge_scatter_kernel<<<gridEdgeWaves, T, 0, stream>>>(
            ei, E, Etot, ebuf, ssum, xl, hout);
    }
    float* hfinal = hB;   // layer 5 writes hB

    // pooling
    hipMemsetAsync(psum,  0, (size_t)G * 64 * 4, stream);
    hipMemsetAsync(pmOrd, 0, (size_t)G * 64 * 4, stream);
    hipMemsetAsync(cnt,   0, (size_t)G * 4, stream);
    pool_kernel<<<gridNodeWaves, T, 0, stream>>>(hfinal, batch, N, psum, pmOrd, cnt);

    // final MLP
    final_mlp_kernel<<<G, 64, 0, stream>>>(
        psum, pmOrd, cnt, prot, fc_w, fc_b, fc2_w, fc2_b, out, G);
}